// ChamferDistance_27058293965199
// MI455X (gfx1250) — compile-verified
//
#include <hip/hip_runtime.h>
#include <hip/hip_bf16.h>

typedef __attribute__((ext_vector_type(2))) float v2f;
typedef __attribute__((ext_vector_type(8))) float v8f;

#define NPTS 4096
#define TILE 16
#define WAVES_PER_BLOCK 8
#define ROWS_PER_BLOCK (WAVES_PER_BLOCK * TILE)   // 128
#define BLOCK_THREADS 256

// One pass: for each point in `src` (rows), find min squared distance to any
// point in `ref` (cols), via V_WMMA_F32_16X16X4_F32 tiles.
//   d2[n,m] = ||x_n||^2 + ||y_m||^2 - 2 <x_n, y_m>
// Factorization that keeps the inner loop lean:
//   A'[n] = (x0, x1, x2, 1)               (||x||^2 added once after the min)
//   B'[m] = (-2y0, -2y1, -2y2, ||y||^2)
//   C     = 0 (inline literal)
//   D[n,m] = ||y||^2 - 2<x,y>;  dist1[n] = max(||x||^2 + min_m D[n,m], 0)
// (clamp commutes with min since max(.,0) is monotone.)
//
// B' is stored in LDS pre-split by lane-half so the per-tile B fragment is a
// single ds_load_b64 with no select VALU:
//   lanes  0-15 carry K=0,1  -> sBlo[m] = (-2y0, -2y1)
//   lanes 16-31 carry K=2,3  -> sBhi[m] = (-2y2, ||y||^2)
__global__ __launch_bounds__(BLOCK_THREADS)
void chamfer_rowmin_kernel(const float* __restrict__ src,   // [B, NPTS, 3] rows
                           const float* __restrict__ ref,   // [B, NPTS, 3] cols
                           float* __restrict__ out_min)     // [B, NPTS] row mins
{
    // 2 x 32 KB = 64 KB of LDS (CDNA5: 320 KB per WGP).
    __shared__ float2 sBlo[NPTS];
    __shared__ float2 sBhi[NPTS];

    const int strips_per_batch = NPTS / ROWS_PER_BLOCK;      // 32
    const int b     = blockIdx.x / strips_per_batch;
    const int strip = blockIdx.x % strips_per_batch;

    const int tid  = threadIdx.x;
    const int lane = tid & 31;
    const int wave = tid >> 5;
    const int half = lane >> 4;     // 0: lanes 0-15 carry K=0,1 / D rows 0-7
                                    // 1: lanes 16-31 carry K=2,3 / D rows 8-15
    const int l16  = lane & 15;

    const float* srcb = src + (size_t)b * NPTS * 3;
    const float* refb = ref + (size_t)b * NPTS * 3;

    // Cooperative preload + transform of the reference cloud for this batch.
    for (int m = tid; m < NPTS; m += BLOCK_THREADS) {
        float y0 = refb[3 * m + 0];
        float y1 = refb[3 * m + 1];
        float y2 = refb[3 * m + 2];
        sBlo[m] = make_float2(-2.0f * y0, -2.0f * y1);
        sBhi[m] = make_float2(-2.0f * y2, y0 * y0 + y1 * y1 + y2 * y2);
    }
    __syncthreads();

    // A fragment (16x4 f32): lane holds row (lane&15); lanes 0-15 carry K=0,1
    // in VGPR0/1, lanes 16-31 carry K=2,3.
    const int row_base = strip * ROWS_PER_BLOCK + wave * TILE;
    const int row      = row_base + l16;
    const float x0 = srcb[3 * row + 0];
    const float x1 = srcb[3 * row + 1];
    const float x2 = srcb[3 * row + 2];
    const float xn = x0 * x0 + x1 * x1 + x2 * x2;

    v2f a;
    a.x = half ? x2   : x0;   // K=2 : K=0
    a.y = half ? 1.0f : x1;   // K=3 : K=1

    // Per-lane base pointer into the half-split B' array (computed once).
    const float2* bbase = (half ? sBhi : sBlo) + l16;

    v8f czero;                // inline-zero accumulator
#pragma unroll
    for (int i = 0; i < 8; ++i) czero[i] = 0.0f;

    v8f rowmin;
#pragma unroll
    for (int i = 0; i < 8; ++i) rowmin[i] = __builtin_inff();

    // Sweep all 256 column tiles of the reference cloud.
#pragma unroll 4
    for (int tm = 0; tm < NPTS; tm += TILE) {
        const float2 bv = bbase[tm];     // ds_load_b64: this lane's B' K-pair
        v2f bb;
        bb.x = bv.x;
        bb.y = bv.y;

        // D = A' x B' + 0  -> 16x16 tile of (||y||^2 - 2<x,y>).
        v8f d = __builtin_amdgcn_wmma_f32_16x16x4_f32(
            /*neg_a=*/false, a, /*neg_b=*/false, bb,
            /*c_mod=*/(short)0, czero, /*reuse_a=*/false, /*reuse_b=*/false);

#pragma unroll
        for (int i = 0; i < 8; ++i)
            rowmin[i] = fminf(rowmin[i], d[i]);
    }

    // Row-min is spread across the 16 lanes of each half: butterfly-reduce
    // within the 16-lane group (masks 1,2,4,8 never cross the half boundary).
#pragma unroll
    for (int s = 1; s < 16; s <<= 1) {
#pragma unroll
        for (int i = 0; i < 8; ++i)
            rowmin[i] = fminf(rowmin[i], __shfl_xor(rowmin[i], s, 32));
    }

    // Gather the ||x||^2 of the 8 rows this half will write. Must run with all
    // lanes active (bpermute reads from disabled lanes return zero).
    float xrow[8];
#pragma unroll
    for (int i = 0; i < 8; ++i)
        xrow[i] = __shfl(xn, half * 8 + i, 32);

    // Lane 0 holds rows 0-7 of the tile, lane 16 holds rows 8-15.
    if (l16 == 0) {
        float* dst = out_min + (size_t)b * NPTS + row_base + half * 8;
#pragma unroll
        for (int i = 0; i < 8; ++i)
            dst[i] = fmaxf(rowmin[i] + xrow[i], 0.0f);   // clamp hoisted here
    }
}

// Sum both min arrays (2 * B * NPTS floats) and scale: result =
// mean(dist1) + mean(dist2) = (sum1 + sum2) / (B*NPTS).
__global__ __launch_bounds__(BLOCK_THREADS)
void chamfer_reduce_kernel(const float* __restrict__ mins, int n,
                           float scale, float* __restrict__ out)
{
    __shared__ float ssum[BLOCK_THREADS];
    float s = 0.0f;
    for (int i = threadIdx.x; i < n; i += BLOCK_THREADS) s += mins[i];
    ssum[threadIdx.x] = s;
    __syncthreads();
    for (int off = BLOCK_THREADS / 2; off > 0; off >>= 1) {
        if (threadIdx.x < off) ssum[threadIdx.x] += ssum[threadIdx.x + off];
        __syncthreads();
    }
    if (threadIdx.x == 0) out[0] = ssum[0] * scale;
}

extern "C" void kernel_launch(void* const* d_in, const int* in_sizes, int n_in,
                              void* d_out, int out_size, void* d_ws, size_t ws_size,
                              hipStream_t stream) {
    (void)in_sizes; (void)n_in; (void)out_size; (void)ws_size;

    const float* xyz1 = (const float*)d_in[0];   // [8, 4096, 3] f32
    const float* xyz2 = (const float*)d_in[1];   // [8, 4096, 3] f32
    float* out = (float*)d_out;                  // scalar f32

    const int B = 8;
    float* dist1 = (float*)d_ws;                 // [B, NPTS]
    float* dist2 = dist1 + (size_t)B * NPTS;     // [B, NPTS]

    const int blocks = B * (NPTS / ROWS_PER_BLOCK);   // 256 blocks, 8 waves each

    // Pass 1: nearest xyz2-point for every xyz1-point.
    chamfer_rowmin_kernel<<<blocks, BLOCK_THREADS, 0, stream>>>(xyz1, xyz2, dist1);
    // Pass 2: symmetric — nearest xyz1-point for every xyz2-point.
    chamfer_rowmin_kernel<<<blocks, BLOCK_THREADS, 0, stream>>>(xyz2, xyz1, dist2);

    const int ntotal = 2 * B * NPTS;                  // 65536
    const float scale = 1.0f / (float)(B * NPTS);     // 1/32768 (each mean)
    chamfer_reduce_kernel<<<1, BLOCK_THREADS, 0, stream>>>(dist1, ntotal, scale, out);
}